// MemoryOptimizedPatchModel_51367808860339
// MI455X (gfx1250) — compile-verified
//
#include <hip/hip_runtime.h>
#include <hip/hip_bf16.h>

#define IMGN  1024
#define PATCH 256
#define TILE  16
#define BATCH 2
#define CHAN  3

typedef float v2f __attribute__((ext_vector_type(2)));
typedef float v8f __attribute__((ext_vector_type(8)));

// ---------------------------------------------------------------- zero d_out
__global__ void zero_kernel(float* __restrict__ out, int n) {
    int i = blockIdx.x * blockDim.x + threadIdx.x;
    if (i < n) out[i] = 0.0f;
}

// ------------------------------------------------ per-patch conv via f32 WMMA
// One wave (32 lanes) handles one 16x16 pixel tile of one (patch, batch).
// K = 28 (= C*3*3 padded from 27), done as 7 chained V_WMMA_F32_16X16X4_F32.
// M = 16 output channels (3 live), N = 16 consecutive pixels in a row.
// The k=27 tap has a zero A-weight, so its B operand is index-clamped (k=26)
// and loaded unconditionally: no EXEC-predicated DS loads in the hot loop.
__global__ __launch_bounds__(32)
void conv_patch_wmma(const float* __restrict__ x,
                     const float* __restrict__ cw,     // [3,3,3,3] OIHW
                     const float* __restrict__ cb,     // [3]
                     const float* __restrict__ wmask,  // [256,256]
                     const int*   __restrict__ locs,   // [N,2]
                     float*       __restrict__ out)    // [B,C,IMG,IMG]
{
    const int lane = threadIdx.x;            // 0..31
    int bid = blockIdx.x;
    const int tilesPerPatch = (PATCH / TILE) * (PATCH / TILE);  // 256
    const int tile = bid % tilesPerPatch;  bid /= tilesPerPatch;
    const int b    = bid % BATCH;
    const int n    = bid / BATCH;
    const int h    = locs[2 * n + 0];
    const int wcol = locs[2 * n + 1];
    const int ty0  = (tile / (PATCH / TILE)) * TILE;   // patch-local tile origin
    const int tx0  = (tile % (PATCH / TILE)) * TILE;

    // 18x18 halo tile per input channel, patch-local zero padding at patch edges.
    __shared__ float sx[CHAN * 18 * 18];
    for (int i = lane; i < CHAN * 18 * 18; i += 32) {
        const int ic = i / 324, rem = i % 324, r = rem / 18, c = rem % 18;
        const int py = ty0 - 1 + r, px = tx0 - 1 + c;
        float v = 0.0f;
        if (py >= 0 && py < PATCH && px >= 0 && px < PATCH)
            v = x[(((size_t)b * CHAN + ic) * IMGN + (h + py)) * IMGN + (wcol + px)];
        sx[i] = v;
    }
    __syncthreads();

    const int oc   = lane & 15;   // A-matrix M index
    const int half = lane >> 4;
    const int n16  = lane & 15;   // B-matrix N index (pixel within row segment)

    // A-matrix (weights): lane -> M = oc, K = 4s + 2*half + j. Zero outside
    // oc<3 / k<27 (this zero is what makes the clamped B load below safe).
    v2f A[7];
#pragma unroll
    for (int s = 0; s < 7; ++s) {
        const int k0 = 4 * s + 2 * half;
        float a0 = 0.0f, a1 = 0.0f;
        if (oc < CHAN && k0     < 27) a0 = cw[oc * 27 + k0];
        if (oc < CHAN && k0 + 1 < 27) a1 = cw[oc * 27 + k0 + 1];
        A[s].x = a0; A[s].y = a1;
    }

    // Per-lane LDS base indices for the 14 im2col taps (row ry adds ry*18).
    int baseIdx[14];
#pragma unroll
    for (int s = 0; s < 7; ++s) {
#pragma unroll
        for (int j = 0; j < 2; ++j) {
            int kk = 4 * s + 2 * half + j;
            if (kk > 26) kk = 26;                 // clamp: weight is 0 there
            const int ic = kk / 9, dy = (kk % 9) / 3, dx = kk % 3;
            baseIdx[2 * s + j] = ic * 324 + dy * 18 + n16 + dx;
        }
    }

    const float b0 = cb[0], b1 = cb[1], b2 = cb[2];

    for (int ry = 0; ry < TILE; ++ry) {
        // Gather all 14 B values first -> one dscnt wait, then dense WMMA chain.
        float bv[14];
        const int rowOff = ry * 18;
#pragma unroll
        for (int t = 0; t < 14; ++t) bv[t] = sx[baseIdx[t] + rowOff];

        v8f acc = {0.f, 0.f, 0.f, 0.f, 0.f, 0.f, 0.f, 0.f};
#pragma unroll
        for (int s = 0; s < 7; ++s) {
            v2f Bv; Bv.x = bv[2 * s]; Bv.y = bv[2 * s + 1];
            acc = __builtin_amdgcn_wmma_f32_16x16x4_f32(
                false, A[s], false, Bv, (short)0, acc, false, false);
        }

        // D layout: lanes 0..15 hold N=pixel, VGPR r holds M=r (oc 0..2 live).
        if (lane < 16) {
            const int py = ty0 + ry, px = tx0 + n16;
            const float m = wmask[py * PATCH + px];
            float v;
            v = acc[0] + b0; v = v > 0.f ? v : 0.f;
            atomicAdd(&out[(((size_t)b * CHAN + 0) * IMGN + (h + py)) * IMGN + (wcol + px)], v * m);
            v = acc[1] + b1; v = v > 0.f ? v : 0.f;
            atomicAdd(&out[(((size_t)b * CHAN + 1) * IMGN + (h + py)) * IMGN + (wcol + px)], v * m);
            v = acc[2] + b2; v = v > 0.f ? v : 0.f;
            atomicAdd(&out[(((size_t)b * CHAN + 2) * IMGN + (h + py)) * IMGN + (wcol + px)], v * m);
        }
    }
}

// -------------------------------------- divide by accumulated window weight
__global__ void normalize_kernel(float* __restrict__ out,
                                 const float* __restrict__ wmask,
                                 const int*   __restrict__ locs, int nLocs)
{
    const int idx = blockIdx.x * blockDim.x + threadIdx.x;
    if (idx >= IMGN * IMGN) return;
    const int y = idx / IMGN, xc = idx % IMGN;
    float ws = 0.0f;
    for (int nn = 0; nn < nLocs; ++nn) {
        const int dy = y - locs[2 * nn + 0];
        const int dx = xc - locs[2 * nn + 1];
        if (dy >= 0 && dy < PATCH && dx >= 0 && dx < PATCH)
            ws += wmask[dy * PATCH + dx];
    }
    const float d = ws + 1e-8f;
    for (int j = 0; j < BATCH * CHAN; ++j) {
        const size_t o = (size_t)j * IMGN * IMGN + idx;
        out[o] = out[o] / d;
    }
}

extern "C" void kernel_launch(void* const* d_in, const int* in_sizes, int n_in,
                              void* d_out, int out_size, void* d_ws, size_t ws_size,
                              hipStream_t stream) {
    const float* x    = (const float*)d_in[0];   // [2,3,1024,1024]
    const float* cw   = (const float*)d_in[1];   // [3,3,3,3]
    const float* cb   = (const float*)d_in[2];   // [3]
    const float* wm   = (const float*)d_in[3];   // [256,256]
    const int*   locs = (const int*)d_in[4];     // [N,2]
    float*       out  = (float*)d_out;           // [2,3,1024,1024]
    const int nLocs = in_sizes[4] / 2;           // 49

    zero_kernel<<<(out_size + 255) / 256, 256, 0, stream>>>(out, out_size);

    const int tilesPerPatch = (PATCH / TILE) * (PATCH / TILE);       // 256
    const int blocks = nLocs * BATCH * tilesPerPatch;                // 25088
    conv_patch_wmma<<<blocks, 32, 0, stream>>>(x, cw, cb, wm, locs, out);

    normalize_kernel<<<(IMGN * IMGN + 255) / 256, 256, 0, stream>>>(out, wm, locs, nLocs);
}